// GCN_15384572854543
// MI455X (gfx1250) — compile-verified
//
#include <hip/hip_runtime.h>
#include <math.h>

typedef __attribute__((ext_vector_type(2))) float v2f;
typedef __attribute__((ext_vector_type(8))) float v8f;

// ---------------------------------------------------------------------------
// Degree / dinv
// ---------------------------------------------------------------------------
__global__ void k_zero(float* __restrict__ p, int n) {
    int i = blockIdx.x * blockDim.x + threadIdx.x;
    if (i < n) p[i] = 0.0f;
}

__global__ void k_deg(const int* __restrict__ dst, float* __restrict__ deg, int E) {
    int e = blockIdx.x * blockDim.x + threadIdx.x;
    if (e < E) unsafeAtomicAdd(&deg[dst[e]], 1.0f);
}

__global__ void k_dinv(float* __restrict__ deg, int n) {
    int i = blockIdx.x * blockDim.x + threadIdx.x;
    if (i < n) deg[i] = rsqrtf(deg[i] + 1.0f);   // +1 self-loop; deg >= 1 always
}

// ---------------------------------------------------------------------------
// fp32 WMMA GEMM: Y[M x (NTILES*16)] = X[M x K] @ W[K x (NTILES*16)] (+ bias)
// One wave computes one 16-row tile across the full N width.
// A-frag (16x4 f32): lane l (l&15 = row), lanehalf picks K+{0,1} vs K+{2,3}.
// B/C/D: row striped across lanes within a VGPR (ISA 7.12.2).
// M must be a multiple of 16 (100000 = 6250*16).
// ---------------------------------------------------------------------------
template <int K, int NTILES>
__global__ void gemm_wmma_f32(const float* __restrict__ X,
                              const float* __restrict__ W,
                              const float* __restrict__ bias,
                              float* __restrict__ Y, int M) {
    const int NCOL = NTILES * 16;
    int waveId = (blockIdx.x * blockDim.x + threadIdx.x) >> 5;
    int lane   = threadIdx.x & 31;
    int nRowTiles = M >> 4;
    if (waveId >= nRowTiles) return;          // wave-uniform: EXEC stays all-1s

    const int row0  = waveId << 4;
    const int r     = lane & 15;              // row within A tile / col within B tile
    const int lh    = lane >> 4;              // lane half
    const int khalf = lh << 1;                // 0 or 2

    const float* xrow = X + (size_t)(row0 + r) * K;

    v8f acc[NTILES] = {};

#pragma unroll
    for (int k0 = 0; k0 < K; k0 += 4) {
        v2f a;
        a.x = xrow[k0 + khalf];
        a.y = xrow[k0 + khalf + 1];
#pragma unroll
        for (int t = 0; t < NTILES; ++t) {
            v2f b;
            int col = t * 16 + r;
            b.x = W[(size_t)(k0 + khalf) * NCOL + col];
            b.y = W[(size_t)(k0 + khalf + 1) * NCOL + col];
            acc[t] = __builtin_amdgcn_wmma_f32_16x16x4_f32(
                false, a, false, b, (short)0, acc[t], false, false);
        }
    }

#pragma unroll
    for (int t = 0; t < NTILES; ++t) {
        int col  = t * 16 + r;
        float bv = bias ? bias[col] : 0.0f;
#pragma unroll
        for (int rr = 0; rr < 8; ++rr) {
            int row = row0 + rr + 8 * lh;     // C/D layout: VGPR rr -> M=rr / rr+8
            Y[(size_t)row * NCOL + col] = acc[t][rr] + bv;
        }
    }
}

// ---------------------------------------------------------------------------
// out[i][f] = g[i][f]*dinv[i]^2 + bias[f]   (self-loop message + bias)
// ---------------------------------------------------------------------------
template <int F>
__global__ void k_init(const float* __restrict__ g, const float* __restrict__ dinv,
                       const float* __restrict__ bias, float* __restrict__ out, int n) {
    int idx = blockIdx.x * blockDim.x + threadIdx.x;
    if (idx >= n * F) return;
    int node = idx / F;
    int f    = idx - node * F;
    float di = dinv[node];
    out[idx] = g[idx] * di * di + bias[f];
}

// ---------------------------------------------------------------------------
// Wave-per-edge scatter: out[dst] += g[src] * dinv[src]*dinv[dst]
// Lane = feature (coalesced 128B gather per wave; f32 L2 atomics for scatter).
// ---------------------------------------------------------------------------
template <int F>
__global__ void edge_scatter(const float* __restrict__ g, const float* __restrict__ dinv,
                             const int* __restrict__ src, const int* __restrict__ dst,
                             float* __restrict__ out, int E) {
    int wid  = (blockIdx.x * blockDim.x + threadIdx.x) >> 5;
    int lane = threadIdx.x & 31;
    if (wid >= E) return;
    int s = src[wid];
    int d = dst[wid];
    float nrm = dinv[s] * dinv[d];
    const float* gs = g + (size_t)s * F;
    float* od       = out + (size_t)d * F;
#pragma unroll
    for (int f = lane; f < F; f += 32)
        unsafeAtomicAdd(&od[f], gs[f] * nrm);
}

__global__ void k_relu(float* __restrict__ p, int n) {
    int i = blockIdx.x * blockDim.x + threadIdx.x;
    if (i < n) p[i] = fmaxf(p[i], 0.0f);
}

// ---------------------------------------------------------------------------
// Row-wise L2 normalize, F = 32: one wave per node, shuffle reduction.
// ---------------------------------------------------------------------------
__global__ void k_norm32(float* __restrict__ out, int n) {
    int wid  = (blockIdx.x * blockDim.x + threadIdx.x) >> 5;
    int lane = threadIdx.x & 31;
    if (wid >= n) return;
    size_t base = (size_t)wid * 32;
    float v  = out[base + lane];
    float ss = v * v;
#pragma unroll
    for (int m = 16; m >= 1; m >>= 1) ss += __shfl_xor(ss, m, 32);
    out[base + lane] = v / fmaxf(sqrtf(ss), 1e-12f);
}

// ---------------------------------------------------------------------------
extern "C" void kernel_launch(void* const* d_in, const int* in_sizes, int n_in,
                              void* d_out, int out_size, void* d_ws, size_t ws_size,
                              hipStream_t stream) {
    const float* x     = (const float*)d_in[0];
    const int*   ei    = (const int*)d_in[1];
    const float* W_pre = (const float*)d_in[2];
    const float* b_pre = (const float*)d_in[3];
    const float* W1    = (const float*)d_in[4];
    const float* b1    = (const float*)d_in[5];
    const float* W2    = (const float*)d_in[6];
    const float* b2    = (const float*)d_in[7];
    const float* W3    = (const float*)d_in[8];
    const float* b3    = (const float*)d_in[9];

    const int n = in_sizes[0] / 128;   // 100000 (multiple of 16)
    const int E = in_sizes[1] / 2;     // 1600000
    const int* src = ei;
    const int* dst = ei + E;

    // Workspace layout: dinv | bufA (n*64) | bufB (n*64)
    char*  ws   = (char*)d_ws;
    size_t o1   = (((size_t)n * 4) + 255) & ~(size_t)255;
    size_t o2   = o1 + ((((size_t)n * 64 * 4) + 255) & ~(size_t)255);
    float* dinv = (float*)ws;
    float* bufA = (float*)(ws + o1);
    float* bufB = (float*)(ws + o2);
    float* out  = (float*)d_out;

    const int T = 256;
    auto cdiv = [](long long a, long long b) { return (int)((a + b - 1) / b); };

    // 1) dinv = rsqrt(deg + 1)
    k_zero<<<cdiv(n, T), T, 0, stream>>>(dinv, n);
    k_deg <<<cdiv(E, T), T, 0, stream>>>(dst, dinv, E);
    k_dinv<<<cdiv(n, T), T, 0, stream>>>(dinv, n);

    const int gemmBlocks = cdiv((long long)(n / 16) * 32, T);
    const int edgeBlocks = cdiv((long long)E * 32, T);

    // 2) h0 = x @ W_pre + b_pre   -> bufA
    gemm_wmma_f32<128, 4><<<gemmBlocks, T, 0, stream>>>(x, W_pre, b_pre, bufA, n);

    // 3) conv1: relu(agg(bufA @ W1) + b1) -> bufA
    gemm_wmma_f32<64, 4><<<gemmBlocks, T, 0, stream>>>(bufA, W1, nullptr, bufB, n);
    k_init<64><<<cdiv((long long)n * 64, T), T, 0, stream>>>(bufB, dinv, b1, bufA, n);
    edge_scatter<64><<<edgeBlocks, T, 0, stream>>>(bufB, dinv, src, dst, bufA, E);
    k_relu<<<cdiv((long long)n * 64, T), T, 0, stream>>>(bufA, n * 64);

    // 4) conv2
    gemm_wmma_f32<64, 4><<<gemmBlocks, T, 0, stream>>>(bufA, W2, nullptr, bufB, n);
    k_init<64><<<cdiv((long long)n * 64, T), T, 0, stream>>>(bufB, dinv, b2, bufA, n);
    edge_scatter<64><<<edgeBlocks, T, 0, stream>>>(bufB, dinv, src, dst, bufA, E);
    k_relu<<<cdiv((long long)n * 64, T), T, 0, stream>>>(bufA, n * 64);

    // 5) conv3 (64 -> 32) into d_out, then L2 normalize
    gemm_wmma_f32<64, 2><<<gemmBlocks, T, 0, stream>>>(bufA, W3, nullptr, bufB, n);
    k_init<32><<<cdiv((long long)n * 32, T), T, 0, stream>>>(bufB, dinv, b3, out, n);
    edge_scatter<32><<<edgeBlocks, T, 0, stream>>>(bufB, dinv, src, dst, out, E);
    k_norm32<<<cdiv((long long)n * 32, T), T, 0, stream>>>(out, n);
}